// HaarDWT_16819091931455
// MI455X (gfx1250) — compile-verified
//
#include <hip/hip_runtime.h>
#include <hip/hip_bf16.h>
#include <stdint.h>

typedef __attribute__((ext_vector_type(4))) float f32x4;

#define H   256
#define W   256
#define H2  128
#define W2  128
#define TH  16            // subband half-rows produced per block (excl. halo)
#define TROWS (TH + 2)    // 18 subband rows incl. +/-1 halo
#define RROWS (2 * TROWS) // 36 raw input rows staged in LDS

__global__ __launch_bounds__(256)
void HaarDWT_16819091931455_kernel(const float* __restrict__ x,
                                   float* __restrict__ out0,
                                   float* __restrict__ out1) {
    __shared__ __align__(16) float raw[RROWS * W];   // 36 KB raw input tile
    __shared__ float sub[4 * TROWS * W2];            // 36 KB subband tile (ll,lh,hl,hh)

    const int tid   = threadIdx.x;
    const int strip = blockIdx.x;      // 0..7
    const int plane = blockIdx.y;      // 0..511  (= n*64 + c)
    const int k0    = strip * TH;      // first half-row of this strip

    const float* pin = x + (size_t)plane * (H * W);

    // ---------- Phase A: async DMA raw tile -> LDS (CDNA5 ASYNCcnt path) ----
    // GVS form: uniform SGPR-pair plane base + 32-bit per-lane byte offset,
    // so no per-lane 64-bit address arithmetic. 2304 float4 chunks; 9/thread.
    const uint32_t raw_base = (uint32_t)(uintptr_t)raw; // low 32b of generic LDS ptr = LDS byte addr
    #pragma unroll
    for (int p = 0; p < 9; ++p) {
        int i  = tid + p * 256;        // float4 index 0..2303
        int t  = i >> 6;               // raw tile row 0..35
        int c4 = (i & 63) << 2;        // col 0,4,...,252
        int hr = k0 - 1 + (t >> 1);    // half-row this raw row feeds
        hr = hr < 0 ? 0 : (hr > H2 - 1 ? H2 - 1 : hr);   // clamp at subband level
        int grow = 2 * hr + (t & 1);
        uint32_t goff = (uint32_t)((grow * W + c4) * 4); // byte offset in plane
        uint32_t lds  = raw_base + (uint32_t)(i << 4);
        asm volatile("global_load_async_to_lds_b128 %0, %1, %2"
                     :: "v"(lds), "v"(goff), "s"(pin) : "memory");
    }
    asm volatile("s_wait_asynccnt 0" ::: "memory");
    __syncthreads();

    // ---------- Phase B: 2x2 Haar analysis -> subband LDS -------------------
    // 18*128 = 2304 coefficients; 9 per thread.
    #pragma unroll
    for (int p = 0; p < 9; ++p) {
        int ci = tid + p * 256;        // 0..2303
        int r  = ci >> 7;              // subband tile row 0..17
        int j  = ci & 127;             // subband col 0..127
        const float2 top = *(const float2*)&raw[(2 * r    ) * W + 2 * j];
        const float2 bot = *(const float2*)&raw[(2 * r + 1) * W + 2 * j];
        float a = top.x, b = top.y, c = bot.x, d = bot.y;
        sub[0 * (TROWS * W2) + r * W2 + j] = 0.5f * (a + b + c + d);  // ll
        sub[1 * (TROWS * W2) + r * W2 + j] = 0.5f * (c + d - a - b);  // lh
        sub[2 * (TROWS * W2) + r * W2 + j] = 0.5f * (b + d - a - c);  // hl
        sub[3 * (TROWS * W2) + r * W2 + j] = 0.5f * (a + d - b - c);  // hh
    }
    __syncthreads();

    // ---------- Phase C: separable 2x bilinear upsample, NT b128 stores -----
    const int n  = plane >> 6;
    const int cc = plane & 63;
    float* llb = out0 + (size_t)plane * (H * W);
    float* hib = out1 + (size_t)(n * 192 + cc) * (H * W);   // lh plane; hl/hh +64/+128 ch

    // 4 subbands x 16 half-rows x 64 col-quads = 4096 micro-tiles; 16 per thread.
    #pragma unroll
    for (int p = 0; p < 16; ++p) {
        int m   = tid + p * 256;       // 0..4095 (s,k uniform per wave)
        int s   = m >> 10;             // subband 0..3
        int rem = m & 1023;
        int k   = rem >> 6;            // half-row in strip 0..15
        int l2  = rem & 63;            // output col quad (cols 4*l2..4*l2+3)
        const float* sb = &sub[s * (TROWS * W2)];
        int c1 = 2 * l2;
        int c0 = (c1 == 0) ? 0 : c1 - 1;          // horizontal edge clamp
        int c2 = c1 + 1;
        int c3 = (c2 == W2 - 1) ? (W2 - 1) : c2 + 1;
        const float* r0 = sb + (k    ) * W2;      // half-row k0+k-1 (halo)
        const float* r1 = sb + (k + 1) * W2;      // half-row k0+k
        const float* r2 = sb + (k + 2) * W2;      // half-row k0+k+1 (halo)
        // vertical blends
        float h0 = 0.25f * r0[c0] + 0.75f * r1[c0];
        float h1 = 0.25f * r0[c1] + 0.75f * r1[c1];
        float h2 = 0.25f * r0[c2] + 0.75f * r1[c2];
        float h3 = 0.25f * r0[c3] + 0.75f * r1[c3];
        float g0 = 0.75f * r1[c0] + 0.25f * r2[c0];
        float g1 = 0.75f * r1[c1] + 0.25f * r2[c1];
        float g2 = 0.75f * r1[c2] + 0.25f * r2[c2];
        float g3 = 0.75f * r1[c3] + 0.25f * r2[c3];
        // horizontal blends -> 4 consecutive output cols, 2 rows
        f32x4 topv = { 0.25f * h0 + 0.75f * h1, 0.75f * h1 + 0.25f * h2,
                       0.25f * h1 + 0.75f * h2, 0.75f * h2 + 0.25f * h3 };
        f32x4 botv = { 0.25f * g0 + 0.75f * g1, 0.75f * g1 + 0.25f * g2,
                       0.25f * g1 + 0.75f * g2, 0.75f * g2 + 0.25f * g3 };
        float* ob = (s == 0) ? llb : (hib + (size_t)(s - 1) * (64 * H * W));
        int oy = 2 * (k0 + k);
        float* op = ob + (size_t)oy * W + 4 * l2;
        __builtin_nontemporal_store(topv, (f32x4*)op);        // row 2k
        __builtin_nontemporal_store(botv, (f32x4*)(op + W));  // row 2k+1
    }
}

extern "C" void kernel_launch(void* const* d_in, const int* in_sizes, int n_in,
                              void* d_out, int out_size, void* d_ws, size_t ws_size,
                              hipStream_t stream) {
    (void)in_sizes; (void)n_in; (void)out_size; (void)d_ws; (void)ws_size;
    const float* x = (const float*)d_in[0];
    float* out0 = (float*)d_out;                               // up(ll): 8*64*256*256
    float* out1 = out0 + (size_t)8 * 64 * 256 * 256;           // concat(lh,hl,hh)
    dim3 grid(H2 / TH, 8 * 64);   // 8 strips x 512 planes
    HaarDWT_16819091931455_kernel<<<grid, 256, 0, stream>>>(x, out0, out1);
}